// Spline_2817498546198
// MI455X (gfx1250) — compile-verified
//
#include <hip/hip_runtime.h>
#include <hip/hip_bf16.h>
#include <stdint.h>

#define ROWS 2048          // B*D = 64*32
#define NK   4096          // knots per row
#define NQ   2048          // queries per row
#define NSEG (NK - 1)      // 4095 segments per row
#define PQ_STRIDE 4096     // padded segment stride (float4 records)

typedef float v2f  __attribute__((ext_vector_type(2)));
typedef float v8f  __attribute__((ext_vector_type(8)));
typedef unsigned int u32x4 __attribute__((ext_vector_type(4)));
typedef int   i32x4 __attribute__((ext_vector_type(4)));
typedef int   i32x8 __attribute__((ext_vector_type(8)));

#if __has_builtin(__builtin_amdgcn_tensor_load_to_lds) && __has_builtin(__builtin_amdgcn_s_wait_tensorcnt)
#define USE_TDM 1
#endif

// ---------------------------------------------------------------------------
// Kernel 1: Thomas solve, one WAVE per row, wave32 parallel scan.
// Forward state s=(alpha,beta,e) evolves by M/m = [[0,u'],[ -l',1 ],[0,b',-l']]
// (structured 3x3: 2x2 block + e-row). cp=alpha/beta, dp=e/beta.
// Backward: x_n = -c_n*x_{n+1} + d_n  -> affine suffix scan.
// All global accesses are coalesced (lane = element within row).
// ---------------------------------------------------------------------------
__global__ __launch_bounds__(256) void spline_thomas(const float* __restrict__ t,
                                                     const float* __restrict__ y,
                                                     float* __restrict__ cs,
                                                     float* __restrict__ dsb,
                                                     float* __restrict__ deriv) {
  const int wave = threadIdx.x >> 5;
  const int lane = threadIdx.x & 31;
  const int row  = blockIdx.x * 8 + wave;
  if (row >= ROWS) return;
  const float* tr = t + (size_t)row * NK;
  const float* yr = y + (size_t)row * NK;
  const size_t ro = (size_t)row * NK;

  float ccp = 0.f, cdp = 0.f;  // forward carry (cp, dp) of previous chunk
  for (int cb = 0; cb < NK; cb += 32) {
    const int n = cb + lane;
    const int nm1 = (n > 0) ? n - 1 : 0;
    const int np1 = (n < NK - 1) ? n + 1 : NK - 1;
    const float tn = tr[n],   yn = yr[n];
    const float tm = tr[nm1], ym = yr[nm1];
    const float tp = tr[np1], yp = yr[np1];
    const float idtm = (n > 0)      ? 1.f / (tn - tm) : 0.f;
    const float idtp = (n < NK - 1) ? 1.f / (tp - tn) : 0.f;
    const float dym = yn - ym, dyp = yp - yn;
    const float l = idtm;                    // dl (0 at n==0)
    const float u = idtp;                    // du (0 at n==N-1)
    const float m = 2.f * (idtm + idtp);     // dm (boundaries collapse correctly)
    const float b = 3.f * (dym * idtm * idtm + dyp * idtp * idtp);
    const float rm = 1.f / m;

    // per-element normalized transfer operator
    float a00 = 0.f,    a01 = u * rm;
    float a10 = -l * rm, a11 = 1.f;
    float p = 0.f, q = b * rm, r = -l * rm;

    // inclusive prefix scan of operator composition (self = later index = left)
#pragma unroll
    for (int off = 1; off < 32; off <<= 1) {
      const float o00 = __shfl_up(a00, off, 32), o01 = __shfl_up(a01, off, 32);
      const float o10 = __shfl_up(a10, off, 32), o11 = __shfl_up(a11, off, 32);
      const float op  = __shfl_up(p,   off, 32), oq  = __shfl_up(q,   off, 32);
      const float orr = __shfl_up(r,   off, 32);
      const bool v = (lane >= off);
      const float n00 = a00 * o00 + a01 * o10;
      const float n01 = a00 * o01 + a01 * o11;
      const float n10 = a10 * o00 + a11 * o10;
      const float n11 = a10 * o01 + a11 * o11;
      const float np_ = p * o00 + q * o10 + r * op;
      const float nq_ = p * o01 + q * o11 + r * oq;
      const float nr_ = r * orr;
      a00 = v ? n00 : a00; a01 = v ? n01 : a01;
      a10 = v ? n10 : a10; a11 = v ? n11 : a11;
      p = v ? np_ : p; q = v ? nq_ : q; r = v ? nr_ : r;
    }
    // apply carry state (ccp, 1, cdp)
    const float alpha = a00 * ccp + a01;
    const float beta  = a10 * ccp + a11;
    const float ev    = p * ccp + q + r * cdp;
    const float rb = 1.f / beta;
    const float cp = alpha * rb, dp = ev * rb;
    cs[ro + n]  = cp;
    dsb[ro + n] = dp;
    ccp = __shfl(cp, 31, 32);
    cdp = __shfl(dp, 31, 32);
  }

  // backward affine suffix scan
  float xc = 0.f;
  for (int cb = NK - 32; cb >= 0; cb -= 32) {
    const int n = cb + lane;
    const float c = cs[ro + n];
    const float d = dsb[ro + n];
    float a = -c, bb = d;   // x_n = a*x_{n+1} + bb
#pragma unroll
    for (int off = 1; off < 32; off <<= 1) {
      const float oa = __shfl_down(a,  off, 32);
      const float ob = __shfl_down(bb, off, 32);
      const bool v = (lane + off < 32);
      const float na = a * oa, nb = a * ob + bb;
      a = v ? na : a; bb = v ? nb : bb;
    }
    const float x = a * xc + bb;
    deriv[ro + n] = x;
    xc = __shfl(x, 0, 32);
  }
}

// ---------------------------------------------------------------------------
// Kernel 2: Bezier basis transform via V_WMMA_F32_16X16X4_F32.
// A (16x4) rows 0..3 hold A_BEZ^T; B (4x16) holds control points c0..c3 for
// 16 segments; D rows 0..3 = p0..p3. Lanes 0-15 store one float4 per segment.
// ---------------------------------------------------------------------------
__device__ __forceinline__ float abez_at(int k, int mrow) {
  // A_mat[M][K] = A_BEZ[K][M] for M<4, else 0
  float r;
  if (k == 0)      r = (mrow == 0) ? 1.f : (mrow == 1) ? -3.f : (mrow == 2) ? 3.f : -1.f;
  else if (k == 1) r = (mrow == 1) ? 3.f : (mrow == 2) ? -6.f : (mrow == 3) ? 3.f : 0.f;
  else if (k == 2) r = (mrow == 2) ? 3.f : (mrow == 3) ? -3.f : 0.f;
  else             r = (mrow == 3) ? 1.f : 0.f;
  return (mrow < 4) ? r : 0.f;
}

__global__ __launch_bounds__(256) void spline_coeffs(const float* __restrict__ t,
                                                     const float* __restrict__ y,
                                                     const float* __restrict__ deriv,
                                                     float4* __restrict__ pq) {
  const int row  = blockIdx.x >> 1;
  const int half = blockIdx.x & 1;
  const int wave = threadIdx.x >> 5;
  const int lane = threadIdx.x & 31;
  const int mrow = lane & 15;
  const int k0   = (lane < 16) ? 0 : 2;
  v2f av;
  av.x = abez_at(k0,     mrow);
  av.y = abez_at(k0 + 1, mrow);
  const size_t ro = (size_t)row * NK;

  for (int it = 0; it < 16; ++it) {
    const int s0 = half * 2048 + (it * 8 + wave) * 16;   // wave-uniform
    const int seg = s0 + mrow;
    const int sc = (seg < NSEG) ? seg : (NSEG - 1);
    const float t0 = t[ro + sc],     t1 = t[ro + sc + 1];
    const float y0 = y[ro + sc],     y1 = y[ro + sc + 1];
    const float d0 = deriv[ro + sc], d1 = deriv[ro + sc + 1];
    const float dt3 = (t1 - t0) * (1.f / 3.f);
    const float c0 = y0;
    const float c1 = __builtin_fmaf(d0,  dt3, y0);
    const float c3 = y1;
    const float c2 = __builtin_fmaf(-d1, dt3, y1);
    v2f bv;
    bv.x = (lane < 16) ? c0 : c2;
    bv.y = (lane < 16) ? c1 : c3;
    v8f acc = {0.f, 0.f, 0.f, 0.f, 0.f, 0.f, 0.f, 0.f};
    // D = A(16x4) x B(4x16) + 0 ; wave-uniform control flow => EXEC all ones
    acc = __builtin_amdgcn_wmma_f32_16x16x4_f32(false, av, false, bv,
                                                (short)0, acc, false, false);
    if (lane < 16 && seg < NSEG) {
      pq[(size_t)row * PQ_STRIDE + seg] = make_float4(acc[0], acc[1], acc[2], acc[3]);
    }
  }
}

// ---------------------------------------------------------------------------
// Kernel 3: evaluation. One block per row; knot row (16 KiB) staged into LDS
// via the Tensor Data Mover; branchless binary search in LDS; single float4
// gather per query; coalesced tq reads and out writes.
// ---------------------------------------------------------------------------
__global__ __launch_bounds__(256) void spline_eval(const float* __restrict__ t,
                                                   const float* __restrict__ tq,
                                                   const float4* __restrict__ pq,
                                                   float* __restrict__ out) {
  __shared__ float s_t[NK];
  const int row = blockIdx.x;
  const float* trow = t + (size_t)row * NK;

#ifdef USE_TDM
  if ((threadIdx.x >> 5) == 0) {       // one wave issues the TDM descriptor
    const uint32_t lds_off = (uint32_t)(uintptr_t)(&s_t[0]);
    const uint64_t ga = (uint64_t)(uintptr_t)trow;
    u32x4 g0;
    g0.x = 1u;                                            // count=1, user mode
    g0.y = lds_off;                                       // lds_addr
    g0.z = (uint32_t)(ga & 0xFFFFFFFFu);                  // global_addr lo
    g0.w = (uint32_t)((ga >> 32) & 0x01FFFFFFu) | (2u << 30); // addr hi | type=2
    i32x8 g1;
    g1[0] = (int)(2u << 16);                  // data_size=4B, wg_mask=0
    g1[1] = (int)((uint32_t)NK << 16);        // tensor_dim0[15:0] @ bits 63:48
    g1[2] = (int)(1u << 16);                  // dim0 hi=0; tensor_dim1=1 @ 95:80
    g1[3] = (int)((uint32_t)NK << 16);        // dim1 hi=0; tile_dim0 @ 127:112
    g1[4] = 0;                                // tile_dim1/2 unused
    g1[5] = NK;                               // tensor_dim0_stride lo
    g1[6] = 0;
    g1[7] = 0;
    const i32x4 z4 = {0, 0, 0, 0};
#if defined(__clang_major__) && (__clang_major__ >= 23)
    const i32x8 z8 = {0, 0, 0, 0, 0, 0, 0, 0};
    __builtin_amdgcn_tensor_load_to_lds(g0, g1, z4, z4, z8, 0);
#else
    __builtin_amdgcn_tensor_load_to_lds(g0, g1, z4, z4, 0);
#endif
    __builtin_amdgcn_s_wait_tensorcnt(0);
  }
#else
  for (int i = threadIdx.x; i < NK; i += 256) s_t[i] = trow[i];
#endif
  __syncthreads();

  const float4* pqr = pq + (size_t)row * PQ_STRIDE;
  const float* tqr  = tq + (size_t)row * NQ;
  float* outr       = out + (size_t)row * NQ;

  for (int qi = threadIdx.x; qi < NQ; qi += 256) {
    const float qv = tqr[qi];
    int pos = 0;                         // searchsorted(side='right')
#pragma unroll
    for (int step = 2048; step >= 1; step >>= 1) {
      const int np = pos + step;
      if (np <= NK && s_t[np - 1] <= qv) pos = np;
    }
    int idx = pos - 1;
    idx = (idx < 0) ? 0 : idx;
    idx = (idx > NK - 2) ? (NK - 2) : idx;
    const float tlo = s_t[idx];
    const float s = (qv - tlo) / (s_t[idx + 1] - tlo);
    const float4 P = pqr[idx];
    const float rv = __builtin_fmaf(
        __builtin_fmaf(__builtin_fmaf(P.w, s, P.z), s, P.y), s, P.x);
    outr[qi] = rv;
  }
}

// ---------------------------------------------------------------------------
extern "C" void kernel_launch(void* const* d_in, const int* in_sizes, int n_in,
                              void* d_out, int out_size, void* d_ws, size_t ws_size,
                              hipStream_t stream) {
  (void)in_sizes; (void)n_in; (void)out_size; (void)ws_size;
  const float* t  = (const float*)d_in[0];
  const float* y  = (const float*)d_in[1];
  const float* tq = (const float*)d_in[2];
  float* out = (float*)d_out;

  // ws layout: [deriv 32MiB][pq 128MiB]; cs/ds scan scratch (64MiB) lives
  // under pq and is dead before spline_coeffs writes pq. Total = 160 MiB.
  float*  deriv = (float*)d_ws;
  float4* pq    = (float4*)(deriv + (size_t)ROWS * NK);
  float*  cs    = (float*)pq;
  float*  dsb   = cs + (size_t)ROWS * NK;

  spline_thomas<<<ROWS / 8, 256, 0, stream>>>(t, y, cs, dsb, deriv);
  spline_coeffs<<<ROWS * 2, 256, 0, stream>>>(t, y, deriv, pq);
  spline_eval<<<ROWS, 256, 0, stream>>>(t, tq, pq, out);
}